// MyGAT_39599598469251
// MI455X (gfx1250) — compile-verified
//
#include <hip/hip_runtime.h>

#define N_HEADS 4
#define HIDC 32
#define DIMF 128   // HEADS*HID
#define NEG_SLOPE 0.2f

typedef float v2f __attribute__((ext_vector_type(2)));
typedef float v8f __attribute__((ext_vector_type(8)));

// ---------- monotonic float<->uint for atomicMax-based segment max ----------
__device__ __forceinline__ unsigned encF(float f) {
  unsigned u = __float_as_uint(f);
  return (u & 0x80000000u) ? ~u : (u | 0x80000000u);
}
__device__ __forceinline__ float decF(unsigned o) {
  return __uint_as_float((o & 0x80000000u) ? (o & 0x7FFFFFFFu) : ~o);
}

// ---------- zero fill ----------
__global__ void k_zero(float* __restrict__ p, long long count) {
  long long tid = (long long)blockIdx.x * blockDim.x + threadIdx.x;
  if (tid < count) p[tid] = 0.0f;
}

// ---------- layer1 projection: one_hot(x) @ W1 == gather rows of W1 ----------
__global__ void k_gather(const int* __restrict__ x, const float* __restrict__ W1,
                         float* __restrict__ h, int n) {
  int tid = blockIdx.x * blockDim.x + threadIdx.x;
  if (tid >= n * DIMF) return;
  int node = tid >> 7, c = tid & (DIMF - 1);
  h[tid] = W1[x[node] * DIMF + c];
}

// ---------- per-node attention dots: as[n,h]=<hproj, att_src>, ad likewise ----
__global__ __launch_bounds__(128) void k_attn_dots(const float* __restrict__ h,
                                                   const float* __restrict__ atts,
                                                   const float* __restrict__ attd,
                                                   float* __restrict__ as_,
                                                   float* __restrict__ ad_) {
  int node = blockIdx.x;
  int head = threadIdx.x >> 5;      // one wave32 per head
  int c = threadIdx.x & 31;
  float v = h[(size_t)node * DIMF + head * HIDC + c];
  float s = v * atts[head * HIDC + c];
  float d = v * attd[head * HIDC + c];
  #pragma unroll
  for (int off = 16; off; off >>= 1) {
    s += __shfl_xor(s, off, 32);
    d += __shfl_xor(d, off, 32);
  }
  if (c == 0) {
    as_[node * N_HEADS + head] = s;
    ad_[node * N_HEADS + head] = d;
  }
}

__device__ __forceinline__ void edge_sd(const int* __restrict__ ei, int e, int E,
                                        int& s, int& d) {
  if (e < E) { s = ei[e]; d = ei[E + e]; }
  else       { s = d = e - E; }         // implicit self-loop
}

// ---------- pass 1: segment max of leaky(e) over dst ----------
__global__ void k_edge_max(const int* __restrict__ ei, const float* __restrict__ as_,
                           const float* __restrict__ ad_, unsigned* __restrict__ emax,
                           int E, int n) {
  int tid = blockIdx.x * blockDim.x + threadIdx.x;
  int tot = (E + n) * N_HEADS;
  if (tid >= tot) return;
  int e = tid >> 2, h = tid & 3;
  int s, d; edge_sd(ei, e, E, s, d);
  float v = as_[s * N_HEADS + h] + ad_[d * N_HEADS + h];
  v = v >= 0.0f ? v : NEG_SLOPE * v;
  atomicMax(&emax[d * N_HEADS + h], encF(v));
}

// ---------- pass 2: denom = segment sum of exp(e - max) ----------
__global__ void k_edge_sum(const int* __restrict__ ei, const float* __restrict__ as_,
                           const float* __restrict__ ad_, const unsigned* __restrict__ emax,
                           float* __restrict__ esum, int E, int n) {
  int tid = blockIdx.x * blockDim.x + threadIdx.x;
  int tot = (E + n) * N_HEADS;
  if (tid >= tot) return;
  int e = tid >> 2, h = tid & 3;
  int s, d; edge_sd(ei, e, E, s, d);
  float v = as_[s * N_HEADS + h] + ad_[d * N_HEADS + h];
  v = v >= 0.0f ? v : NEG_SLOPE * v;
  float ex = __expf(v - decF(emax[d * N_HEADS + h]));
  atomicAdd(&esum[d * N_HEADS + h], ex);
}

// ---------- pass 3: out[dst] += alpha * hproj[src]  (per edge-channel) -------
__global__ void k_edge_scatter(const int* __restrict__ ei, const float* __restrict__ as_,
                               const float* __restrict__ ad_, const unsigned* __restrict__ emax,
                               const float* __restrict__ esum, const float* __restrict__ hin,
                               float* __restrict__ hout, int E, int n) {
  int tid = blockIdx.x * blockDim.x + threadIdx.x;
  int tot = (E + n) * DIMF;
  if (tid >= tot) return;
  int e = tid >> 7, c = tid & (DIMF - 1), h = c >> 5;
  int s, d; edge_sd(ei, e, E, s, d);
  __builtin_prefetch(&hin[(size_t)s * DIMF + c], 0, 0);   // global_prefetch_b8
  float v = as_[s * N_HEADS + h] + ad_[d * N_HEADS + h];
  v = v >= 0.0f ? v : NEG_SLOPE * v;
  float m = decF(emax[d * N_HEADS + h]);
  float alpha = __expf(v - m) / esum[d * N_HEADS + h];
  atomicAdd(&hout[(size_t)d * DIMF + c], alpha * hin[(size_t)s * DIMF + c]);
}

// ---------- bias (+optional relu) in place ----------
__global__ void k_bias(float* __restrict__ hv, const float* __restrict__ b, int n, int relu) {
  int tid = blockIdx.x * blockDim.x + threadIdx.x;
  if (tid >= n * DIMF) return;
  int c = tid & (DIMF - 1);
  float v = hv[tid] + b[c];
  if (relu) v = fmaxf(v, 0.0f);
  hv[tid] = v;
}

// ---------- O[N,128] = H[N,128] @ W[128,128] via V_WMMA_F32_16X16X4_F32 ------
// block = 256 threads = 8 waves; block owns one 16-row tile, wave w owns 16-col
// tile w. A tile (16x128, 8KB) staged in LDS and shared by all 8 waves; W (64KB)
// is hot in L2 across all 6250 blocks.
__global__ __launch_bounds__(256) void k_gemm128(const float* __restrict__ H,
                                                 const float* __restrict__ W,
                                                 float* __restrict__ O) {
  __shared__ float sA[16 * DIMF];
  int rowBase = blockIdx.x * 16;
  for (int i = threadIdx.x; i < 16 * DIMF; i += 256) {
    int r = i >> 7, c = i & (DIMF - 1);
    sA[i] = H[(size_t)(rowBase + r) * DIMF + c];
  }
  __syncthreads();

  int wave = threadIdx.x >> 5;
  int lane = threadIdx.x & 31;
  int colBase = wave * 16;
  int mn = lane & 15;               // A: M index / B: N index
  int kh = (lane >> 4) * 2;         // K sub-offset: lanes 0-15 -> K0,K1 ; 16-31 -> K2,K3
  v8f acc = {};
  #pragma unroll 4
  for (int k = 0; k < DIMF; k += 4) {
    v2f a, b;
    a.x = sA[mn * DIMF + k + kh + 0];
    a.y = sA[mn * DIMF + k + kh + 1];
    b.x = W[(size_t)(k + kh + 0) * DIMF + colBase + mn];
    b.y = W[(size_t)(k + kh + 1) * DIMF + colBase + mn];
    acc = __builtin_amdgcn_wmma_f32_16x16x4_f32(false, a, false, b,
                                                (short)0, acc, false, false);
  }
  int rowOff = (lane >> 4) * 8;     // C/D: VGPR v -> M=v (lanes 0-15), M=v+8 (16-31)
  #pragma unroll
  for (int v = 0; v < 8; ++v)
    O[(size_t)(rowBase + rowOff + v) * DIMF + colBase + mn] = acc[v];
}

// ---------- global mean pool (sum + count) ----------
__global__ void k_pool(const float* __restrict__ h, const int* __restrict__ batch,
                       float* __restrict__ pooled, float* __restrict__ cnt, int n) {
  int tid = blockIdx.x * blockDim.x + threadIdx.x;
  if (tid >= n * DIMF) return;
  int node = tid >> 7, c = tid & (DIMF - 1);
  int g = batch[node];
  atomicAdd(&pooled[g * DIMF + c], h[tid]);
  if (c == 0) atomicAdd(&cnt[g], 1.0f);
}

// ---------- classifier: out[g,o] = (pooled[g]/cnt[g]) @ lin_w + lin_b --------
__global__ __launch_bounds__(32) void k_classify(const float* __restrict__ pooled,
                                                 const float* __restrict__ cnt,
                                                 const float* __restrict__ lw,
                                                 const float* __restrict__ lb,
                                                 float* __restrict__ out, int OUTC) {
  int g = blockIdx.x;
  int o = threadIdx.x;
  if (o >= OUTC) return;
  float inv = 1.0f / fmaxf(cnt[g], 1.0f);
  float acc = lb[o];
  #pragma unroll 8
  for (int k = 0; k < DIMF; ++k)
    acc += pooled[g * DIMF + k] * inv * lw[k * OUTC + o];
  out[g * OUTC + o] = acc;
}

extern "C" void kernel_launch(void* const* d_in, const int* in_sizes, int n_in,
                              void* d_out, int out_size, void* d_ws, size_t ws_size,
                              hipStream_t stream) {
  const int*   x    = (const int*)  d_in[0];
  const int*   ei   = (const int*)  d_in[1];
  const int*   batch= (const int*)  d_in[2];
  const float* W1   = (const float*)d_in[3];
  const float* as1  = (const float*)d_in[4];
  const float* ad1  = (const float*)d_in[5];
  const float* b1   = (const float*)d_in[6];
  const float* W3   = (const float*)d_in[7];
  const float* as3  = (const float*)d_in[8];
  const float* ad3  = (const float*)d_in[9];
  const float* b3   = (const float*)d_in[10];
  const float* lw   = (const float*)d_in[11];
  const float* lb   = (const float*)d_in[12];
  float* out = (float*)d_out;

  const int N = in_sizes[0];
  const int E = in_sizes[1] / 2;
  const int OUTC = 10;
  const int G = out_size / OUTC;

  // workspace carve-up
  float*    hA    = (float*)d_ws;                 // N*128  (proj features)
  float*    hB    = hA + (size_t)N * DIMF;        // N*128  (aggregation / next input)
  float*    as_   = hB + (size_t)N * DIMF;        // N*4
  float*    ad_   = as_ + (size_t)N * N_HEADS;    // N*4
  unsigned* emax  = (unsigned*)(ad_ + (size_t)N * N_HEADS); // N*4
  float*    esum  = (float*)(emax + (size_t)N * N_HEADS);   // N*4
  float*    pooled= esum + (size_t)N * N_HEADS;   // G*128
  float*    cnt   = pooled + (size_t)G * DIMF;    // G

  const int TB = 256;
  long long ncELL = (long long)N * DIMF;
  int ncBlocks  = (int)((ncELL + TB - 1) / TB);
  int ehBlocks  = ((E + N) * N_HEADS + TB - 1) / TB;
  long long ecELL = (long long)(E + N) * DIMF;
  int ecBlocks  = (int)((ecELL + TB - 1) / TB);
  int znBlocks  = (2 * N * N_HEADS + TB - 1) / TB;

  // ===== layer 1 =====
  k_gather<<<ncBlocks, TB, 0, stream>>>(x, W1, hA, N);
  k_attn_dots<<<N, 128, 0, stream>>>(hA, as1, ad1, as_, ad_);
  k_zero<<<ncBlocks, TB, 0, stream>>>(hB, ncELL);
  k_zero<<<znBlocks, TB, 0, stream>>>((float*)emax, 2LL * N * N_HEADS); // emax+esum
  k_edge_max    <<<ehBlocks, TB, 0, stream>>>(ei, as_, ad_, emax, E, N);
  k_edge_sum    <<<ehBlocks, TB, 0, stream>>>(ei, as_, ad_, emax, esum, E, N);
  k_edge_scatter<<<ecBlocks, TB, 0, stream>>>(ei, as_, ad_, emax, esum, hA, hB, E, N);
  k_bias<<<ncBlocks, TB, 0, stream>>>(hB, b1, N, /*relu=*/1);

  // ===== layer 2: WMMA projection hA = hB @ W3 =====
  k_gemm128<<<N / 16, 256, 0, stream>>>(hB, W3, hA);
  k_attn_dots<<<N, 128, 0, stream>>>(hA, as3, ad3, as_, ad_);
  k_zero<<<ncBlocks, TB, 0, stream>>>(hB, ncELL);
  k_zero<<<znBlocks, TB, 0, stream>>>((float*)emax, 2LL * N * N_HEADS);
  k_edge_max    <<<ehBlocks, TB, 0, stream>>>(ei, as_, ad_, emax, E, N);
  k_edge_sum    <<<ehBlocks, TB, 0, stream>>>(ei, as_, ad_, emax, esum, E, N);
  k_edge_scatter<<<ecBlocks, TB, 0, stream>>>(ei, as_, ad_, emax, esum, hA, hB, E, N);
  k_bias<<<ncBlocks, TB, 0, stream>>>(hB, b3, N, /*relu=*/0);

  // ===== pool + classify =====
  int zpBlocks = (G * DIMF + G + TB - 1) / TB;
  k_zero<<<zpBlocks, TB, 0, stream>>>(pooled, (long long)G * DIMF + G);
  k_pool<<<ncBlocks, TB, 0, stream>>>(hB, batch, pooled, cnt, N);
  k_classify<<<G, 32, 0, stream>>>(pooled, cnt, lw, lb, out, OUTC);
}